// RotateWeights_6030134083831
// MI455X (gfx1250) — compile-verified
//
#include <hip/hip_runtime.h>

// out = H * W^T * H / n  (n = 8192, Sylvester Hadamard, orthonormal FWHT twice)
// Implemented as two bandwidth-bound passes:
//   pass1: mid = rowFWHT(W)            (coalesced rows -> coalesced rows, into d_ws)
//   pass2: out = rowFWHT(mid^T)        (transpose-read via LDS, coalesced row writes)
// Each row's FWHT = 3x dense H16 contractions on the matrix pipe
// (v_wmma_f32_16x16x4_f32, K=16 via 4 chained WMMAs) + one H2 butterfly stage.
// Requires ws_size >= 8192*8192*4 = 256 MB.

typedef __attribute__((ext_vector_type(2))) float v2f;
typedef __attribute__((ext_vector_type(8))) float v8f;

constexpr int   N         = 8192;
constexpr int   ROWS      = 8;          // rows (or columns) per workgroup
constexpr int   ROWSTRIDE = N + 8;      // LDS row pitch in floats (bank-conflict pad)
constexpr int   THREADS   = 256;        // 8 wave32 -> one wave per row
constexpr float INV_SQRT_N = 0.011048543456039806f;  // 1/sqrt(8192)

__device__ __forceinline__ float hsign(int r, int c) {
  // Sylvester H16 entry: (-1)^popcount(r&c)
  return (__popc(r & c) & 1) ? -1.0f : 1.0f;
}

// D = M * H16   (A = 16x16 data tile from LDS, row pitch rs; B = H16 chunks)
__device__ __forceinline__ v8f tile_mul_right_H(const float* t, int rs,
                                                const v2f* h, int half, int l15) {
  v8f acc = {};
#pragma unroll
  for (int kk = 0; kk < 4; ++kk) {
    const int ka = 4 * kk + 2 * half;          // A: lanes 0-15 K=4kk+{0,1}; 16-31 K=4kk+{2,3}
    v2f a;
    a.x = t[l15 * rs + ka];
    a.y = t[l15 * rs + ka + 1];
    acc = __builtin_amdgcn_wmma_f32_16x16x4_f32(false, a, false, h[kk],
                                                (short)0, acc, false, false);
  }
  return acc;
}

// D = H16 * M   (A = H16 chunks; B = 16x16 data tile from LDS, row pitch rs)
__device__ __forceinline__ v8f tile_mul_left_H(const float* t, int rs,
                                               const v2f* h, int half, int l15) {
  v8f acc = {};
#pragma unroll
  for (int kk = 0; kk < 4; ++kk) {
    const int ka = 4 * kk + 2 * half;          // B: lanes 0-15 K=4kk+{0,1}; 16-31 K=4kk+{2,3}
    v2f b;
    b.x = t[ka * rs + l15];
    b.y = t[(ka + 1) * rs + l15];
    acc = __builtin_amdgcn_wmma_f32_16x16x4_f32(false, h[kk], false, b,
                                                (short)0, acc, false, false);
  }
  return acc;
}

// C/D layout: VGPR e -> row (e | e+8), col = lane&15
__device__ __forceinline__ void store_tile(float* t, int rs, int half, int l15, v8f acc) {
#pragma unroll
  for (int e = 0; e < 8; ++e)
    t[(e + 8 * half) * rs + l15] = acc[e];
}

template <int TRANS>
__global__ __launch_bounds__(THREADS)
void fwht_rows(const float* __restrict__ in, float* __restrict__ out) {
  extern __shared__ float lds[];
  const int tid   = threadIdx.x;
  const int w     = tid >> 5;        // wave id = LDS row
  const int lane  = tid & 31;
  const int half  = lane >> 4;
  const int l15   = lane & 15;
  const int base0 = blockIdx.x * ROWS;
  float* row = lds + w * ROWSTRIDE;

  if (TRANS == 0) {
    // each wave streams its own row: fully coalesced 512B/wave loads
    const float4* src = (const float4*)(in + (size_t)(base0 + w) * N);
    float4* dstl = (float4*)row;
#pragma unroll 4
    for (int i = lane; i < N / 4; i += 32) dstl[i] = src[i];
  } else {
    // cooperative transposed read: LDS row j <- column (base0+j) of `in`
    const int j0 = (tid & 1) * 4;
    for (int k = tid >> 1; k < N; k += THREADS / 2) {
      float4 v = *(const float4*)(in + (size_t)k * N + base0 + j0);
      lds[(j0 + 0) * ROWSTRIDE + k] = v.x;
      lds[(j0 + 1) * ROWSTRIDE + k] = v.y;
      lds[(j0 + 2) * ROWSTRIDE + k] = v.z;
      lds[(j0 + 3) * ROWSTRIDE + k] = v.w;
    }
    __syncthreads();
  }

  // H16 operand chunks (A and B layouts coincide since H is symmetric)
  v2f h[4];
#pragma unroll
  for (int kk = 0; kk < 4; ++kk) {
    const int ka = 4 * kk + 2 * half;
    h[kk].x = hsign(l15, ka);
    h[kk].y = hsign(l15, ka + 1);
  }

  // H8192 = H2 (x) H16 (x) H16 (x) H16 on index bits [c|d|b|j]
  // stage 1: contract j (contiguous 16s): tile t covers rows i=t*16..+15, D = M*H
  for (int t = 0; t < 32; ++t) {
    float* tp = row + t * 256;
    v8f acc = tile_mul_right_H(tp, 16, h, half, l15);
    store_tile(tp, 16, half, l15, acc);
  }
  // stage 2: contract b (stride 16): same tile memory, D = H*M
  for (int t = 0; t < 32; ++t) {
    float* tp = row + t * 256;
    v8f acc = tile_mul_left_H(tp, 16, h, half, l15);
    store_tile(tp, 16, half, l15, acc);
  }
  // stage 3: contract d (stride 256): tiles fixed (c,b), D = H*M
  for (int u = 0; u < 32; ++u) {
    float* tp = row + (u >> 4) * 4096 + (u & 15) * 16;
    v8f acc = tile_mul_left_H(tp, 256, h, half, l15);
    store_tile(tp, 256, half, l15, acc);
  }
  // stage 4: H2 butterfly over c (stride 4096), fold in 1/sqrt(n)
  for (int i = lane * 4; i < 4096; i += 128) {
    float4 a = *(float4*)(row + i);
    float4 b = *(float4*)(row + i + 4096);
    float4 p, m;
    p.x = (a.x + b.x) * INV_SQRT_N;  m.x = (a.x - b.x) * INV_SQRT_N;
    p.y = (a.y + b.y) * INV_SQRT_N;  m.y = (a.y - b.y) * INV_SQRT_N;
    p.z = (a.z + b.z) * INV_SQRT_N;  m.z = (a.z - b.z) * INV_SQRT_N;
    p.w = (a.w + b.w) * INV_SQRT_N;  m.w = (a.w - b.w) * INV_SQRT_N;
    *(float4*)(row + i)        = p;
    *(float4*)(row + i + 4096) = m;
  }

  // each wave stores its own row: fully coalesced
  float4* dst = (float4*)(out + (size_t)(base0 + w) * N);
  const float4* srcl = (const float4*)row;
#pragma unroll 4
  for (int i = lane; i < N / 4; i += 32) dst[i] = srcl[i];
}

extern "C" void kernel_launch(void* const* d_in, const int* in_sizes, int n_in,
                              void* d_out, int out_size, void* d_ws, size_t ws_size,
                              hipStream_t stream) {
  (void)in_sizes; (void)n_in; (void)out_size; (void)ws_size;
  const float* W  = (const float*)d_in[0];
  float* out      = (float*)d_out;
  float* mid      = (float*)d_ws;   // needs 8192*8192*4 = 256 MB

  const dim3 grid(N / ROWS), block(THREADS);
  const size_t shmem = (size_t)ROWS * ROWSTRIDE * sizeof(float);  // ~256 KB dynamic LDS

  fwht_rows<0><<<grid, block, shmem, stream>>>(W, mid);   // mid = W*H/sqrt(n)
  fwht_rows<1><<<grid, block, shmem, stream>>>(mid, out); // out = H*W^T*H/n
}